// Generator_33552284516717
// MI455X (gfx1250) — compile-verified
//
#include <hip/hip_runtime.h>

// ---------------- problem constants ----------------
#define HID    150
#define HIDP   152        // HID padded to multiple of 4
#define FEATN  2
#define OUTN   5
#define LATN   100
#define ZK     105        // LATENT + OUT
#define KXY    8          // FEAT+OUT (=7) padded to 8
#define NTILE  10         // ceil(HID/16) hidden tiles
#define NUNITS 20         // 2 dirs * NTILE
#define GATE_ROWS 160     // HID padded to 160 rows per gate

// ---------------- workspace layout (floats) ----------------
#define OFF_WIH0 0
#define SZ_WIH0  (2*4*GATE_ROWS*KXY)        // 10240
#define OFF_WHH0 (OFF_WIH0 + SZ_WIH0)
#define SZ_WHH0  (2*4*GATE_ROWS*HIDP)       // 194560
#define OFF_WIH1 (OFF_WHH0 + SZ_WHH0)
#define SZ_WIH1  (2*4*GATE_ROWS*(2*HIDP))   // 389120
#define OFF_WHH1 (OFF_WIH1 + SZ_WIH1)
#define SZ_WHH1  (2*4*GATE_ROWS*HIDP)       // 194560
#define WS_TOTAL (OFF_WHH1 + SZ_WHH1)       // 788480 floats ~ 3.0 MB

typedef float v2f __attribute__((ext_vector_type(2)));
typedef float v8f __attribute__((ext_vector_type(8)));

__device__ __forceinline__ v8f wmma4(v2f a, v2f b, v8f c) {
  // V_WMMA_F32_16X16X4_F32 : D = A(16x4) * B(4x16) + C(16x16)
  return __builtin_amdgcn_wmma_f32_16x16x4_f32(false, a, false, b, (short)0, c,
                                               false, false);
}

__device__ __forceinline__ float sigf(float x) {
  return 1.0f / (1.0f + __expf(-x));
}

// ---------------- weight repack: zero-padded per-gate layout ----------------
__global__ void prep_weights(const float* __restrict__ Wih0,
                             const float* __restrict__ Whh0,
                             const float* __restrict__ Wih1,
                             const float* __restrict__ Whh1,
                             float* __restrict__ ws) {
  long i = (long)blockIdx.x * blockDim.x + threadIdx.x;
  if (i < SZ_WIH0) {                       // [2][4][160][8] <- (2,600,7)
    int k = i & 7;
    int r = (i >> 3) % GATE_ROWS;
    int q = ((i >> 3) / GATE_ROWS) & 3;
    int d = (int)(i / (8 * GATE_ROWS * 4));
    float v = (r < HID && k < (FEATN + OUTN)) ? Wih0[(d*600 + q*HID + r)*7 + k] : 0.f;
    ws[OFF_WIH0 + i] = v; return;
  }
  i -= SZ_WIH0;
  if (i < SZ_WHH0) {                       // [2][4][160][152] <- (2,600,150)
    int k = i % HIDP;
    int r = (int)(i / HIDP) % GATE_ROWS;
    int q = (int)(i / (HIDP * GATE_ROWS)) & 3;
    int d = (int)(i / (HIDP * GATE_ROWS * 4));
    float v = (r < HID && k < HID) ? Whh0[(d*600 + q*HID + r)*HID + k] : 0.f;
    ws[OFF_WHH0 + i] = v; return;
  }
  i -= SZ_WHH0;
  if (i < SZ_WIH1) {                       // [2][4][160][304] <- (2,600,300)
    int k = i % (2 * HIDP);
    int r = (int)(i / (2 * HIDP)) % GATE_ROWS;
    int q = (int)(i / (2 * HIDP * GATE_ROWS)) & 3;
    int d = (int)(i / (2 * HIDP * GATE_ROWS * 4));
    int half = k / HIDP, kr = k % HIDP;
    float v = (r < HID && kr < HID) ? Wih1[(d*600 + q*HID + r)*300 + half*HID + kr] : 0.f;
    ws[OFF_WIH1 + i] = v; return;
  }
  i -= SZ_WIH1;
  if (i < SZ_WHH1) {                       // [2][4][160][152] <- (2,600,150)
    int k = i % HIDP;
    int r = (int)(i / HIDP) % GATE_ROWS;
    int q = (int)(i / (HIDP * GATE_ROWS)) & 3;
    int d = (int)(i / (HIDP * GATE_ROWS * 4));
    float v = (r < HID && k < HID) ? Whh1[(d*600 + q*HID + r)*HID + k] : 0.f;
    ws[OFF_WHH1 + i] = v; return;
  }
}

// ---------------- main persistent-per-block LSTM kernel ----------------
__global__ __launch_bounds__(256)
void lstm_gen_kernel(const float* __restrict__ noise,
                     const float* __restrict__ targets,
                     const float* __restrict__ masks,
                     const int*   __restrict__ lengths,
                     const int*   __restrict__ tsp,
                     const float* __restrict__ b_l0,
                     const float* __restrict__ b_l1,
                     const float* __restrict__ W_proj,
                     const float* __restrict__ b_proj,
                     const float* __restrict__ W_inith,
                     const float* __restrict__ b_inith,
                     const float* __restrict__ W_inittok,
                     const float* __restrict__ b_inittok,
                     const float* __restrict__ wsf,
                     float* __restrict__ out,
                     int Btot) {
  const int blk  = blockIdx.x;
  const int b0   = blk * 16;
  const int tid  = threadIdx.x;
  const int lane = tid & 31;
  const int w    = tid >> 5;         // wave id (8 waves)
  const int n    = lane & 15;        // column within tile / batch row for A
  const int hi   = lane >> 4;        // half-wave select
  const int Bq8  = Btot >> 3;

  __shared__ float sH0[2][16][HIDP];
  __shared__ float sH1[2][16][HIDP];
  __shared__ float sXY[16][KXY];
  __shared__ float sZ[32][ZK];

  // ---- stage the 32 z-rows this block needs ----
  // rows 0..15 : z[q*(B/8) + 2*blk + (row&1)], q = row>>1  (for h0c0 init)
  // rows 16..31: z[b0 + row-16]                             (for x init)
  for (int i = tid; i < 32 * ZK; i += 256) {
    int row = i / ZK, k = i - row * ZK;
    int zb = (row < 16) ? ((row >> 1) * Bq8 + 2 * blk + (row & 1)) : (b0 + row - 16);
    float v;
    if (k < LATN) v = noise[zb * LATN + k];
    else {
      int j = k - LATN;
      v = (j < FEATN) ? targets[zb * FEATN + j]
                      : masks[zb * (OUTN - FEATN) + (j - FEATN)];
    }
    sZ[row][k] = v;
  }
  __syncthreads();

  // ---- init h0, h1 (row-major padded; pad cols stay 0 forever) ----
  for (int i = tid; i < 2 * 2 * 16 * HIDP; i += 256) {
    int k   = i % HIDP;
    int m   = (i / HIDP) & 15;
    int d   = (i / (HIDP * 16)) & 1;
    int lay = i / (HIDP * 16 * 2);
    float v = 0.f;
    if (k < HID) {
      int q    = lay * 2 + d;                 // h_l0 -> q=d, h_l1 -> q=2+d
      int colG = (m & 7) * HID + k;
      v = b_inith[colG];
      const float* wr = W_inith + colG * ZK;
      const float* zr = sZ[q * 2 + (m >> 3)];
      for (int kk = 0; kk < ZK; ++kk) v += zr[kk] * wr[kk];
    }
    if (lay == 0) sH0[d][m][k] = v; else sH1[d][m][k] = v;
  }

  // ---- init xy = [x(2), y(5), pad] ----
  for (int i = tid; i < 16 * KXY; i += 256) {
    int m = i / KXY, k = i - m * KXY;
    float v = 0.f;
    if (k < FEATN) {
      v = b_inittok[k];
      const float* wr = W_inittok + k * ZK;
      const float* zr = sZ[16 + m];
      for (int kk = 0; kk < ZK; ++kk) v += zr[kk] * wr[kk];
    } else if (k < FEATN + OUTN) {
      int j = k - FEATN;
      v = (j < FEATN) ? targets[(b0 + m) * FEATN + j]
                      : masks[(b0 + m) * (OUTN - FEATN) + (j - FEATN)];
    }
    sXY[m][k] = v;
  }

  // ---- init c state: kept in registers in D-fragment layout ----
  float cr0[3][8], cr1[3][8];
#pragma unroll
  for (int s = 0; s < 3; ++s) {
    int u = w + s * 8;
    if (u < NUNITS) {
      int d = u / NTILE, jt = u - d * NTILE;
      int col = jt * 16 + n;
#pragma unroll
      for (int r = 0; r < 8; ++r) {
        int m = r + 8 * hi;
        float v0 = 0.f, v1 = 0.f;
        if (col < HID) {
          int colG = (m & 7) * HID + col;
          const float* wr = W_inith + colG * ZK;
          v0 = b_inith[colG]; v1 = v0;
          const float* z0 = sZ[(4 + d) * 2 + (m >> 3)];   // c_l0 -> q=4+d
          const float* z1 = sZ[(6 + d) * 2 + (m >> 3)];   // c_l1 -> q=6+d
          for (int kk = 0; kk < ZK; ++kk) { v0 += z0[kk] * wr[kk]; v1 += z1[kk] * wr[kk]; }
        }
        cr0[s][r] = v0; cr1[s][r] = v1;
      }
    } else {
#pragma unroll
      for (int r = 0; r < 8; ++r) { cr0[s][r] = 0.f; cr1[s][r] = 0.f; }
    }
  }
  __syncthreads();

  const float* Wih0p = wsf + OFF_WIH0;
  const float* Whh0p = wsf + OFF_WHH0;
  const float* Wih1p = wsf + OFF_WIH1;
  const float* Whh1p = wsf + OFF_WHH1;
  const v8f vzero = {0.f,0.f,0.f,0.f,0.f,0.f,0.f,0.f};
  const int ts = tsp[0];
  const int k2 = 2 * hi;

  for (int t = 0; t < ts; ++t) {
    v8f acc[3][4];

    // ================= Layer 0 GEMM =================
#pragma unroll
    for (int s = 0; s < 3; ++s) {
      int u = w + s * 8;
      if (u >= NUNITS) continue;
      int d = u / NTILE, jt = u - d * NTILE;
      int rowbase = jt * 16 + n;
#pragma unroll
      for (int q = 0; q < 4; ++q) acc[s][q] = vzero;
      // phase 1: inp = [x|y], K=8
#pragma unroll
      for (int k0 = 0; k0 < KXY; k0 += 4) {
        v2f a = *(const v2f*)&sXY[n][k0 + k2];
#pragma unroll
        for (int q = 0; q < 4; ++q) {
          const float* bp = Wih0p + ((d*4 + q) * GATE_ROWS + rowbase) * KXY + k0 + k2;
          acc[s][q] = wmma4(a, *(const v2f*)bp, acc[s][q]);
        }
      }
      // phase 2: h0[d], K=152
      for (int k0 = 0; k0 < HIDP; k0 += 4) {
        v2f a = *(const v2f*)&sH0[d][n][k0 + k2];
#pragma unroll
        for (int q = 0; q < 4; ++q) {
          const float* bp = Whh0p + ((d*4 + q) * GATE_ROWS + rowbase) * HIDP + k0 + k2;
          acc[s][q] = wmma4(a, *(const v2f*)bp, acc[s][q]);
        }
      }
    }
    __syncthreads();

    // ================= Layer 0 elementwise =================
#pragma unroll
    for (int s = 0; s < 3; ++s) {
      int u = w + s * 8;
      if (u >= NUNITS) continue;
      int d = u / NTILE, jt = u - d * NTILE;
      int col = jt * 16 + n;
      if (col < HID) {
        const float* bb = b_l0 + d * 600;
#pragma unroll
        for (int r = 0; r < 8; ++r) {
          int m = r + 8 * hi;
          float iv = acc[s][0][r] + bb[col];
          float fv = acc[s][1][r] + bb[HID + col];
          float gv = acc[s][2][r] + bb[2 * HID + col];
          float ov = acc[s][3][r] + bb[3 * HID + col];
          float c  = sigf(fv) * cr0[s][r] + sigf(iv) * tanhf(gv);
          cr0[s][r] = c;
          sH0[d][m][col] = sigf(ov) * tanhf(c);
        }
      }
    }
    __syncthreads();

    // ================= Layer 1 GEMM =================
#pragma unroll
    for (int s = 0; s < 3; ++s) {
      int u = w + s * 8;
      if (u >= NUNITS) continue;
      int d = u / NTILE, jt = u - d * NTILE;
      int rowbase = jt * 16 + n;
#pragma unroll
      for (int q = 0; q < 4; ++q) acc[s][q] = vzero;
      // phase 1: l1in = [h0_fwd | h0_bwd], K=304 (two padded halves)
      for (int k0 = 0; k0 < 2 * HIDP; k0 += 4) {
        int kk   = k0 + k2;
        int dsrc = (kk >= HIDP) ? 1 : 0;
        v2f a = *(const v2f*)&sH0[dsrc][n][kk - dsrc * HIDP];
#pragma unroll
        for (int q = 0; q < 4; ++q) {
          const float* bp = Wih1p + ((d*4 + q) * GATE_ROWS + rowbase) * (2 * HIDP) + kk;
          acc[s][q] = wmma4(a, *(const v2f*)bp, acc[s][q]);
        }
      }
      // phase 2: h1[d], K=152
      for (int k0 = 0; k0 < HIDP; k0 += 4) {
        v2f a = *(const v2f*)&sH1[d][n][k0 + k2];
#pragma unroll
        for (int q = 0; q < 4; ++q) {
          const float* bp = Whh1p + ((d*4 + q) * GATE_ROWS + rowbase) * HIDP + k0 + k2;
          acc[s][q] = wmma4(a, *(const v2f*)bp, acc[s][q]);
        }
      }
    }
    __syncthreads();

    // ================= Layer 1 elementwise =================
#pragma unroll
    for (int s = 0; s < 3; ++s) {
      int u = w + s * 8;
      if (u >= NUNITS) continue;
      int d = u / NTILE, jt = u - d * NTILE;
      int col = jt * 16 + n;
      if (col < HID) {
        const float* bb = b_l1 + d * 600;
#pragma unroll
        for (int r = 0; r < 8; ++r) {
          int m = r + 8 * hi;
          float iv = acc[s][0][r] + bb[col];
          float fv = acc[s][1][r] + bb[HID + col];
          float gv = acc[s][2][r] + bb[2 * HID + col];
          float ov = acc[s][3][r] + bb[3 * HID + col];
          float c  = sigf(fv) * cr1[s][r] + sigf(iv) * tanhf(gv);
          cr1[s][r] = c;
          sH1[d][m][col] = sigf(ov) * tanhf(c);
        }
      }
    }
    __syncthreads();

    // ================= projection + output + next x =================
    if (w == 0) {
      int m = lane >> 1, f = lane & 1;
      float a = b_proj[f];
      const float* wp = W_proj + f * 300;
      for (int k = 0; k < HID; ++k) a += sH1[0][m][k] * wp[k];
      for (int k = 0; k < HID; ++k) a += sH1[1][m][k] * wp[HID + k];
      sXY[m][f] = a;                               // x for next step
      int bb = b0 + m;
      out[((long)bb * ts + t) * FEATN + f] = (t < lengths[bb]) ? a : 0.f;
    }
    __syncthreads();
  }
}

// ---------------- launcher ----------------
extern "C" void kernel_launch(void* const* d_in, const int* in_sizes, int n_in,
                              void* d_out, int out_size, void* d_ws, size_t ws_size,
                              hipStream_t stream) {
  const float* noise     = (const float*)d_in[0];
  const float* targets   = (const float*)d_in[1];
  const float* masks     = (const float*)d_in[2];
  const int*   lengths   = (const int*)d_in[3];
  const int*   tsp       = (const int*)d_in[4];
  const float* Wih_l0    = (const float*)d_in[5];
  const float* Whh_l0    = (const float*)d_in[6];
  const float* b_l0      = (const float*)d_in[7];
  const float* Wih_l1    = (const float*)d_in[8];
  const float* Whh_l1    = (const float*)d_in[9];
  const float* b_l1      = (const float*)d_in[10];
  const float* W_proj    = (const float*)d_in[11];
  const float* b_proj    = (const float*)d_in[12];
  const float* W_inith   = (const float*)d_in[13];
  const float* b_inith   = (const float*)d_in[14];
  const float* W_inittok = (const float*)d_in[15];
  const float* b_inittok = (const float*)d_in[16];
  float* wsf = (float*)d_ws;
  int Btot = in_sizes[0] / LATN;   // 2048

  prep_weights<<<(WS_TOTAL + 255) / 256, 256, 0, stream>>>(
      Wih_l0, Whh_l0, Wih_l1, Whh_l1, wsf);

  lstm_gen_kernel<<<Btot / 16, 256, 0, stream>>>(
      noise, targets, masks, lengths, tsp,
      b_l0, b_l1, W_proj, b_proj,
      W_inith, b_inith, W_inittok, b_inittok,
      wsf, (float*)d_out, Btot);
}